// Emu3VisionVQVectorQuantizer_11166914969804
// MI455X (gfx1250) — compile-verified
//
#include <hip/hip_runtime.h>

typedef __attribute__((ext_vector_type(2))) float v2f;
typedef __attribute__((ext_vector_type(8))) float v8f;

namespace {
constexpr int   KCODES = 32768;   // codebook size
constexpr int   NVEC   = 8192;    // B*T*H*W = 1*2*64*64
constexpr int   HWSZ   = 4096;    // H*W
constexpr float BETA   = 0.1f;
}

// ---------------------------------------------------------------------------
// Kernel 1: precompute ||e_k||^2 for every code; thread 0 zeroes the loss slot.
// ---------------------------------------------------------------------------
__global__ __launch_bounds__(256) void vq_enorm_kernel(const float* __restrict__ E,
                                                       float* __restrict__ eNorm,
                                                       float* __restrict__ qloss) {
    const int k = blockIdx.x * 256 + threadIdx.x;          // KCODES % 256 == 0
    const float4 e = *(const float4*)(E + k * 4);
    eNorm[k] = e.x * e.x + e.y * e.y + e.z * e.z + e.w * e.w;
    if (k == 0) *qloss = 0.0f;
}

// ---------------------------------------------------------------------------
// Kernel 2: WMMA distance tiles + running argmin.
// One wave32 owns 16 query rows; loops over 32768 codes in chunks of 16.
// ---------------------------------------------------------------------------
__global__ __launch_bounds__(256) void vq_argmin_kernel(const float* __restrict__ x,
                                                        const float* __restrict__ E,
                                                        const float* __restrict__ eNorm,
                                                        int*   __restrict__ codes_i,
                                                        float* __restrict__ codes_f) {
    const int lane = threadIdx.x & 31;
    const int wave = (blockIdx.x * blockDim.x + threadIdx.x) >> 5;
    const int tileBase = wave * 16;          // first query row of this tile
    const int half = lane >> 4;              // 0 -> K=0,1 ; 1 -> K=2,3
    const int ln   = lane & 15;

    // ---- A fragment (16x4 f32, ISA layout): row M = ln, channels 2*half, 2*half+1
    const int row = tileBase + ln;
    const int bt  = row >> 12;               // row / 4096
    const int hw  = row & (HWSZ - 1);
    const float* xrow = x + bt * (4 * HWSZ) + hw;   // channel stride = 4096
    v2f a;
    a.x = xrow[(half * 2 + 0) * HWSZ];
    a.y = xrow[(half * 2 + 1) * HWSZ];

    float best[8];
    int   bidx[8];
#pragma unroll
    for (int r = 0; r < 8; ++r) { best[r] = 3.0e38f; bidx[r] = 0; }

    for (int base = 0; base < KCODES; base += 16) {
        if ((base & 1023) == 0)  // warm L2/WGP$ ahead on the cold first sweep
            __builtin_prefetch(E + (base + 1024) * 4, 0, 1);

        const int code = base + ln;          // this lane's column n = ln
        // ---- B fragment (4x16 f32): column n = code, K-halves split by lane half
        const float2 e = *(const float2*)(E + code * 4 + half * 2);
        v2f b; b.x = e.x; b.y = e.y;
        const float en = eNorm[code];

        v8f c = {};
        // D = A x B (+0): 16x16 tile of fp32 dot products in one instruction
        c = __builtin_amdgcn_wmma_f32_16x16x4_f32(
                /*neg_a=*/false, a, /*neg_b=*/false, b,
                /*c_mod=*/(short)0, c, /*reuse_a=*/false, /*reuse_b=*/false);

#pragma unroll
        for (int r = 0; r < 8; ++r) {
            // distance surrogate: ||e||^2 - 2 x.e  (||x||^2 constant per row)
            const float d = fmaf(-2.0f, c[r], en);
            const bool lt = d < best[r];
            best[r] = lt ? d : best[r];
            bidx[r] = lt ? code : bidx[r];
        }
    }

    // ---- argmin butterfly within each 16-lane group (xor 1,2,4,8 stays in group)
#pragma unroll
    for (int off = 1; off < 16; off <<= 1) {
#pragma unroll
        for (int r = 0; r < 8; ++r) {
            const float od = __shfl_xor(best[r], off, 32);
            const int   oi = __shfl_xor(bidx[r], off, 32);
            const bool take = (od < best[r]) || ((od == best[r]) && (oi < bidx[r]));
            best[r] = take ? od : best[r];
            bidx[r] = take ? oi : bidx[r];
        }
    }

    // D layout: lanes 0-15 hold rows 0..7 (VGPR r), lanes 16-31 hold rows 8..15
    if (ln == 0) {
#pragma unroll
        for (int r = 0; r < 8; ++r) {
            const int m = r + half * 8;
            codes_i[tileBase + m] = bidx[r];
            codes_f[tileBase + m] = (float)bidx[r];
        }
    }
}

// ---------------------------------------------------------------------------
// Kernel 3: gather z_q = E[code] into (B,T,C,H,W) layout + scalar loss.
// forward(z_q_st) == z_q ; qloss == (1+BETA) * mean((z_q - x)^2)
// ---------------------------------------------------------------------------
__global__ __launch_bounds__(256) void vq_gather_kernel(const float* __restrict__ x,
                                                        const float* __restrict__ E,
                                                        const int* __restrict__ codes_i,
                                                        float* __restrict__ zq,
                                                        float* __restrict__ qloss) {
    const int n = blockIdx.x * 256 + threadIdx.x;          // NVEC % 256 == 0
    const int code = codes_i[n];
    const int bt = n >> 12;
    const int hw = n & (HWSZ - 1);
    const float4 e = *(const float4*)(E + code * 4);
    const float ev[4] = { e.x, e.y, e.z, e.w };

    float err = 0.0f;
#pragma unroll
    for (int c = 0; c < 4; ++c) {
        const int off = bt * (4 * HWSZ) + c * HWSZ + hw;
        const float xv = x[off];
        zq[off] = ev[c];
        const float d = ev[c] - xv;
        err = fmaf(d, d, err);
    }
    err *= (1.0f + BETA) / (float)(4 * NVEC);

    // wave32 reduce, then one atomic per wave
#pragma unroll
    for (int off = 16; off >= 1; off >>= 1)
        err += __shfl_xor(err, off, 32);
    if ((threadIdx.x & 31) == 0)
        atomicAdd(qloss, err);
}

// ---------------------------------------------------------------------------
// d_out layout (flat, return order): z_q_st [32768 f32] | qloss [1 f32] |
//                                    codes  [8192, stored as float values]
// d_ws layout: eNorm [32768 f32] | codes_i [8192 i32]
// ---------------------------------------------------------------------------
extern "C" void kernel_launch(void* const* d_in, const int* in_sizes, int n_in,
                              void* d_out, int out_size, void* d_ws, size_t ws_size,
                              hipStream_t stream) {
    const float* x = (const float*)d_in[0];   // (1,2,4,64,64) f32
    const float* E = (const float*)d_in[1];   // (32768,4) f32

    float* out     = (float*)d_out;
    float* zq      = out;                     // 32768
    float* qloss   = out + 32768;             // 1
    float* codes_f = out + 32769;             // 8192

    float* eNorm   = (float*)d_ws;            // 32768 f32
    int*   codes_i = (int*)((float*)d_ws + KCODES);   // 8192 i32

    vq_enorm_kernel<<<KCODES / 256, 256, 0, stream>>>(E, eNorm, qloss);

    // 512 tiles of 16 rows -> 512 waves -> 16384 threads
    vq_argmin_kernel<<<(NVEC / 16) * 32 / 256, 256, 0, stream>>>(
        x, E, eNorm, codes_i, codes_f);

    vq_gather_kernel<<<NVEC / 256, 256, 0, stream>>>(x, E, codes_i, zq, qloss);
}